// MemoryCell_21449066676319
// MI455X (gfx1250) — compile-verified
//
#include <hip/hip_runtime.h>
#include <stdint.h>

// Problem constants (fixed by the reference: B=32, T=128, d=128, units=256)
#define D_     128
#define UNITS_ 256
#define B_     32
#define T_     128
#define TC_    16                  // timesteps staged per LDS tile
#define BPB_   2                   // batches per block
#define NTHR_  (BPB_ * D_)         // 256 threads = 8 wave32
#define NUNF_  6                   // ODE_UNFOLDS
#define EPS_   1e-8f
#define LOG2E_ 1.4426950408889634f
#define TILE_FLOATS_ (BPB_ * TC_ * D_)   // 4096 floats = 16 KB per buffer

__device__ __forceinline__ float rcp_fast(float x) {
    return __builtin_amdgcn_rcpf(x);           // v_rcp_f32
}

// sigmoid((v - m) * g) with precomputed A = -g*log2e, C = m*g*log2e:
//   = 1 / (1 + exp2(fma(v, A, C)))            -> v_fma, v_exp, v_add, v_rcp
__device__ __forceinline__ float sigmoid_pre(float v, float A, float C) {
    return rcp_fast(1.0f + __builtin_amdgcn_exp2f(__builtin_fmaf(v, A, C)));
}

__device__ __forceinline__ float div_fast(float n, float d) {
    // rcp + one Newton-Raphson refinement, then multiply (~1 ulp)
    float r = __builtin_amdgcn_rcpf(d);
    r = __builtin_fmaf(__builtin_fmaf(-d, r, 1.0f), r, r);
    return n * r;
}

extern "C" __global__ __launch_bounds__(NTHR_)
void ltc_memcell_kernel(
    const float* __restrict__ inputs,                        // (B,T,D)
    const float* __restrict__ input_w,  const float* __restrict__ input_b,   // (D,)
    const float* __restrict__ output_w, const float* __restrict__ output_b,  // (D,)
    const float* __restrict__ gleak,    const float* __restrict__ vleak,     // (256,)
    const float* __restrict__ cm,                                            // (256,)
    const float* __restrict__ w,   const float* __restrict__ mu,             // (256,256)
    const float* __restrict__ sg,  const float* __restrict__ er,             // (256,256)
    const float* __restrict__ sw,  const float* __restrict__ smu,            // (128,256)
    const float* __restrict__ ssg, const float* __restrict__ ser,            // (128,256)
    float* __restrict__ out)                                                 // (B,D)
{
    __shared__ float tile[2][TILE_FLOATS_];

    const int tid = threadIdx.x;
    const int bl  = tid >> 7;          // local batch (0..1)
    const int j   = tid & (D_ - 1);    // pair index / motor unit
    const int b   = blockIdx.x * BPB_ + bl;
    const int j2  = j + D_;            // partner (memory) unit

    // ---------------- per-pair parameters, resident in VGPRs ----------------
    const float iw    = input_w[j];
    const float ibias = input_b[j];
    const float cmt1  = cm[j]  * (float)NUNF_;   // cm * ode_unfolds
    const float cmt2  = cm[j2] * (float)NUNF_;
    const float gl1   = gleak[j],  gl2 = gleak[j2];
    const float gv1   = gl1 * vleak[j];
    const float gv2   = gl2 * vleak[j2];

    // On-mask recurrent synapses: column j <- rows {j, j+D}; column j2 <- rows {j2, j}
    const int i_ss1 = j  * UNITS_ + j;     // self  -> unit j
    const int i_cs1 = j2 * UNITS_ + j;     // cross -> unit j
    const int i_ss2 = j2 * UNITS_ + j2;    // self  -> unit j2
    const int i_cs2 = j  * UNITS_ + j2;    // cross -> unit j2

    // Fold (v - mu)*sigma into a single FMA feeding v_exp (base 2):
    //   A = -sigma*log2e (coeff of v),  C = mu*sigma*log2e (constant)
    const float w_ss1 = w[i_ss1], A_ss1 = -sg[i_ss1]*LOG2E_, C_ss1 = mu[i_ss1]*sg[i_ss1]*LOG2E_, e_ss1 = er[i_ss1];
    const float w_cs1 = w[i_cs1], A_cs1 = -sg[i_cs1]*LOG2E_, C_cs1 = mu[i_cs1]*sg[i_cs1]*LOG2E_, e_cs1 = er[i_cs1];
    const float w_ss2 = w[i_ss2], A_ss2 = -sg[i_ss2]*LOG2E_, C_ss2 = mu[i_ss2]*sg[i_ss2]*LOG2E_, e_ss2 = er[i_ss2];
    const float w_cs2 = w[i_cs2], A_cs2 = -sg[i_cs2]*LOG2E_, C_cs2 = mu[i_cs2]*sg[i_cs2]*LOG2E_, e_cs2 = er[i_cs2];

    // On-mask sensory synapses: sensory channel j -> units {j, j2}
    const int i_s1 = j * UNITS_ + j;
    const int i_s2 = j * UNITS_ + j2;
    const float sw1 = sw[i_s1], sA1 = -ssg[i_s1]*LOG2E_, sC1 = smu[i_s1]*ssg[i_s1]*LOG2E_, se1 = ser[i_s1];
    const float sw2 = sw[i_s2], sA2 = -ssg[i_s2]*LOG2E_, sC2 = smu[i_s2]*ssg[i_s2]*LOG2E_, se2 = ser[i_s2];

    // Timestep-invariant partial sums
    const float num_base1 = gv1;                       // + wnum_s per timestep
    const float num_base2 = gv2;
    const float den_base1 = cmt1 + gl1 + EPS_;         // + wden_s per timestep
    const float den_base2 = cmt2 + gl2 + EPS_;

    // addrspace(3) pointer truncates to the workgroup-relative LDS byte offset
    const uint32_t lds_base = (uint32_t)(uintptr_t)(&tile[0][0]);

    // --------- async global -> LDS staging of one input tile (CDNA5) ---------
    auto issue_tile = [&](int buf, int t0) {
        const int per = NTHR_ * 4;                     // 1024 floats per issue
        #pragma unroll
        for (int k = 0; k < TILE_FLOATS_ / per; ++k) { // 4 issues of B128/lane
            int fb  = k * per + tid * 4;               // flat float idx in tile
            int bb  = fb >> 11;                        // TC_*D_ = 2048
            int rem = fb & (TC_ * D_ - 1);
            int tl  = rem >> 7;
            int jj  = rem & (D_ - 1);
            const float* gp = inputs +
                ((size_t)(blockIdx.x * BPB_ + bb) * T_ + (size_t)(t0 + tl)) * D_ + jj;
            uint32_t la = lds_base + (uint32_t)((buf * TILE_FLOATS_ + fb) * 4);
            uint64_t ga = (uint64_t)(uintptr_t)gp;
            asm volatile("global_load_async_to_lds_b128 %0, %1, off"
                         :: "v"(la), "v"(ga) : "memory");
        }
    };

    // Prefetch tile 0
    issue_tile(0, 0);

    float v1 = 0.0f, v2 = 0.0f;
    int cur = 0;

    for (int t0 = 0; t0 < T_; t0 += TC_) {
        // Current tile's async loads done (per-wave), then make visible block-wide
        asm volatile("s_wait_asynccnt 0x0" ::: "memory");
        __syncthreads();

        // Kick off next tile into the other buffer while we compute this one.
        // Safe: writes into buffer `cur` only happen after the *next* top
        // barrier, which all waves reach only after finishing reading `cur`.
        if (t0 + TC_ < T_) issue_tile(cur ^ 1, t0 + TC_);

        const float* xt = &tile[cur][bl * (TC_ * D_) + j];

        #pragma unroll 2
        for (int tl = 0; tl < TC_; ++tl) {
            // affine input mapping + sensory activations (once per timestep)
            const float x   = __builtin_fmaf(xt[tl * D_], iw, ibias);
            const float sa1 = sw1 * sigmoid_pre(x, sA1, sC1);
            const float sa2 = sw2 * sigmoid_pre(x, sA2, sC2);
            // fold sensory sums + leak terms into unfold-invariant constants
            const float gvw1 = __builtin_fmaf(sa1, se1, num_base1);  // gl*vl + wnum_s
            const float gvw2 = __builtin_fmaf(sa2, se2, num_base2);
            const float glw1 = den_base1 + sa1;                      // cmt+gl+eps+wden_s
            const float glw2 = den_base2 + sa2;

            #pragma unroll
            for (int u = 0; u < NUNF_; ++u) {
                // synapse conductances from *old* presynaptic voltages
                const float a1 = w_ss1 * sigmoid_pre(v1, A_ss1, C_ss1);
                const float c1 = w_cs1 * sigmoid_pre(v2, A_cs1, C_cs1);
                const float a2 = w_ss2 * sigmoid_pre(v2, A_ss2, C_ss2);
                const float c2 = w_cs2 * sigmoid_pre(v1, A_cs2, C_cs2);

                const float num1 = __builtin_fmaf(c1, e_cs1,
                                   __builtin_fmaf(a1, e_ss1,
                                   __builtin_fmaf(cmt1, v1, gvw1)));
                const float den1 = glw1 + a1 + c1;
                const float num2 = __builtin_fmaf(c2, e_cs2,
                                   __builtin_fmaf(a2, e_ss2,
                                   __builtin_fmaf(cmt2, v2, gvw2)));
                const float den2 = glw2 + a2 + c2;

                const float nv1 = div_fast(num1, den1);
                const float nv2 = div_fast(num2, den2);
                v1 = nv1;
                v2 = nv2;
            }
        }
        cur ^= 1;
    }

    // motor neurons = first D units, affine output mapping
    out[(size_t)b * D_ + j] = __builtin_fmaf(v1, output_w[j], output_b[j]);
}

extern "C" void kernel_launch(void* const* d_in, const int* in_sizes, int n_in,
                              void* d_out, int out_size, void* d_ws, size_t ws_size,
                              hipStream_t stream) {
    (void)in_sizes; (void)n_in; (void)d_ws; (void)ws_size; (void)out_size;

    const float* inputs   = (const float*)d_in[0];
    const float* input_w  = (const float*)d_in[1];
    const float* input_b  = (const float*)d_in[2];
    const float* output_w = (const float*)d_in[3];
    const float* output_b = (const float*)d_in[4];
    const float* gleak    = (const float*)d_in[5];
    const float* vleak    = (const float*)d_in[6];
    const float* cm       = (const float*)d_in[7];
    const float* w        = (const float*)d_in[8];
    const float* mu       = (const float*)d_in[9];
    const float* sigma    = (const float*)d_in[10];
    const float* erev     = (const float*)d_in[11];
    const float* sw       = (const float*)d_in[12];
    const float* smu      = (const float*)d_in[13];
    const float* ssg      = (const float*)d_in[14];
    const float* ser      = (const float*)d_in[15];

    dim3 grid(B_ / BPB_);   // 16 blocks
    dim3 block(NTHR_);      // 256 threads (8 wave32)
    hipLaunchKernelGGL(ltc_memcell_kernel, grid, block, 0, stream,
                       inputs, input_w, input_b, output_w, output_b,
                       gleak, vleak, cm, w, mu, sigma, erev,
                       sw, smu, ssg, ser, (float*)d_out);
}